// LovaszSoftmax_38920993636661
// MI455X (gfx1250) — compile-verified
//
#include <hip/hip_runtime.h>
#include <hip/hip_bf16.h>

typedef float v2f __attribute__((ext_vector_type(2)));
typedef float v8f __attribute__((ext_vector_type(8)));

#define NCLS 6
#define BATCH 8
#define HW (512 * 512)               // 2^18
#define NPIX (BATCH * HW)            // 2^21
#define NBINS (1 << 20)              // counting-sort bins over float-bit key
#define BINS_PER_BLOCK 2048          // 8 waves * 256 bins
#define NSCAN_BLOCKS (NBINS / BINS_PER_BLOCK)   // 512

// ---------------------------------------------------------------- zero scratch
__global__ void lovasz_zero_kernel(float* __restrict__ w, long long total,
                                   float* __restrict__ loss) {
  long long i = (long long)blockIdx.x * blockDim.x + threadIdx.x;
  long long stride = (long long)gridDim.x * blockDim.x;
  for (; i < total; i += stride) w[i] = 0.0f;
  if (blockIdx.x == 0 && threadIdx.x < NCLS) loss[threadIdx.x] = 0.0f;
}

// -------------------------------------------------- softmax + histogram fill
// hist layout per class c: [cnt[NBINS] | fg[NBINS] | errSum[NBINS]]
__global__ void lovasz_fill_kernel(const float* __restrict__ logits,
                                   const int* __restrict__ labels,
                                   float* __restrict__ hist) {
  int n = blockIdx.x * blockDim.x + threadIdx.x;
  if (n >= NPIX) return;
  int b = n >> 18;              // HW == 2^18
  int hw = n & (HW - 1);
  const float* lg = logits + (size_t)b * NCLS * HW + hw;

  float l[NCLS];
  float m = -1e30f;
#pragma unroll
  for (int c = 0; c < NCLS; ++c) {
    l[c] = lg[(size_t)c * HW];
    m = fmaxf(m, l[c]);
  }
  float s = 0.0f;
#pragma unroll
  for (int c = 0; c < NCLS; ++c) {
    l[c] = expf(l[c] - m);
    s += l[c];
  }
  float inv = 1.0f / s;

  int lab = labels[n];
  if (lab == 0) return;         // IGNORE_INDEX: contributes nothing anywhere

#pragma unroll
  for (int c = 0; c < NCLS; ++c) {
    float p = l[c] * inv;
    float fg = (lab == c) ? 1.0f : 0.0f;
    float err = fabsf(fg - p);                       // in [0,1]
    // monotone: larger err -> smaller bin index (descending sort order)
    unsigned bin = (0x3F800000u - __float_as_uint(err)) >> 10;
    if (bin >= NBINS) bin = NBINS - 1;
    float* base = hist + (size_t)c * 3 * NBINS;
    atomicAdd(base + bin, 1.0f);                     // valid count (exact f32)
    if (fg > 0.0f) atomicAdd(base + NBINS + bin, 1.0f);
    atomicAdd(base + 2 * NBINS + bin, err);          // per-bin error sum
  }
}

// ------------------------------------------- WMMA block-level inclusive scan
// Each wave scans 256 bins as a 16x16 column-major tile: D = L * X with
// v_wmma_f32_16x16x4_f32 chained over K (L = lower-triangular ones, exact).
__global__ void lovasz_scan_block_kernel(const float* __restrict__ in,
                                         float* __restrict__ out,
                                         float* __restrict__ blockSums) {
  __shared__ float wsums[8];
  const int lane = threadIdx.x & 31;
  const int wave = threadIdx.x >> 5;
  const int col = lane & 15;      // N index (B/C/D lane mapping)
  const int half = lane >> 4;     // K-half for A/B operands
  const int base = blockIdx.x * BINS_PER_BLOCK + wave * 256;

  v8f acc = {};
#pragma unroll
  for (int t = 0; t < 4; ++t) {
    const int k0 = 4 * t + half * 2;
    v2f a, b;
    // A = L[:, k0:k0+2] for this lane (M == lane&15): L[m][k] = (k <= m)
    a.x = (k0 <= col) ? 1.0f : 0.0f;
    a.y = (k0 + 1 <= col) ? 1.0f : 0.0f;
    // B = X rows k0..k0+1 for column `col` (X[k][n] = in[base + n*16 + k])
    b.x = in[base + col * 16 + k0];
    b.y = in[base + col * 16 + k0 + 1];
    acc = __builtin_amdgcn_wmma_f32_16x16x4_f32(false, a, false, b, (short)0,
                                                acc, false, false);
  }
  // column totals = row 15 of D: reg 7, lanes 16..31 (N = lane-16)
  float tcol = __shfl(acc[7], 16 + col, 32);
  // inclusive shuffle-scan across the 16 columns (both halves identical)
  float sc = tcol;
#pragma unroll
  for (int off = 1; off < 16; off <<= 1) {
    float v = __shfl(sc, col - off, 32);
    if (col >= off) sc += v;
  }
  float excl = sc - tcol;
  float waveTotal = __shfl(sc, 15, 32);

  if (lane == 0) wsums[wave] = waveTotal;
  __syncthreads();
  float woff = 0.0f;
#pragma unroll
  for (int w2 = 0; w2 < 8; ++w2)
    if (w2 < wave) woff += wsums[w2];

  float add = excl + woff;
#pragma unroll
  for (int j = 0; j < 8; ++j) {
    int row = j + half * 8;     // D reg j: M = j (lanes 0-15) / j+8 (16-31)
    out[base + col * 16 + row] = acc[j] + add;
  }
  if (threadIdx.x == 0) {
    float bt = 0.0f;
#pragma unroll
    for (int w2 = 0; w2 < 8; ++w2) bt += wsums[w2];
    blockSums[blockIdx.x] = bt;
  }
}

// ------------------------------------- top-level scan of 512 block totals
__global__ void lovasz_scan_top_kernel(const float* __restrict__ bsC,
                                       const float* __restrict__ bsF,
                                       float* __restrict__ boC,
                                       float* __restrict__ boF,
                                       float* __restrict__ gts, int cls) {
  __shared__ float sC[NSCAN_BLOCKS];
  __shared__ float sF[NSCAN_BLOCKS];
  int tid = threadIdx.x;
  float c0 = bsC[tid], f0 = bsF[tid];
  sC[tid] = c0;
  sF[tid] = f0;
  __syncthreads();
  for (int off = 1; off < NSCAN_BLOCKS; off <<= 1) {
    float vc = 0.0f, vf = 0.0f;
    if (tid >= off) { vc = sC[tid - off]; vf = sF[tid - off]; }
    __syncthreads();
    sC[tid] += vc;
    sF[tid] += vf;
    __syncthreads();
  }
  boC[tid] = sC[tid] - c0;   // exclusive block offsets
  boF[tid] = sF[tid] - f0;
  if (tid == NSCAN_BLOCKS - 1) gts[cls] = sF[tid];   // total foreground count
}

// ----------------------------- per-bin Lovasz contribution (telescoped run)
__global__ void lovasz_finalize_kernel(const float* __restrict__ hist,
                                       const float* __restrict__ scC,
                                       const float* __restrict__ scF,
                                       const float* __restrict__ boC,
                                       const float* __restrict__ boF,
                                       const float* __restrict__ gts,
                                       float* __restrict__ loss, int cls) {
  int bin = blockIdx.x * blockDim.x + threadIdx.x;
  const float* base = hist + (size_t)cls * 3 * NBINS;
  float contrib = 0.0f;
  float n = base[bin];
  if (n > 0.0f) {
    float f = base[NBINS + bin];
    float es = base[2 * NBINS + bin];
    int sb = bin >> 11;                       // 2048 bins per scan block
    float i1 = scC[bin] + boC[sb];            // inclusive cum-valid
    float F1 = scF[bin] + boF[sb];            // inclusive cum-fg
    float i0 = i1 - n;
    float F0 = F1 - f;
    float G = gts[cls];
    // J(i,F) = 1 - (G-F)/(G+i-F); denom >= max(G,i) >= 1 when i >= 1
    float J1 = 1.0f - (G - F1) / (G + i1 - F1);
    float J0 = (i0 > 0.5f) ? (1.0f - (G - F0) / (G + i0 - F0)) : 0.0f;
    contrib = (es / n) * (J1 - J0);           // run of equal errors telescopes
  }
  __shared__ float red[256];
  red[threadIdx.x] = contrib;
  __syncthreads();
  for (int off = 128; off > 0; off >>= 1) {
    if (threadIdx.x < off) red[threadIdx.x] += red[threadIdx.x + off];
    __syncthreads();
  }
  if (threadIdx.x == 0) atomicAdd(&loss[cls], red[0]);
}

// ------------------------------------------------------------ final combine
__global__ void lovasz_combine_kernel(const float* __restrict__ gts,
                                      const float* __restrict__ loss,
                                      float* __restrict__ out) {
  float s = 0.0f, np = 0.0f;
  for (int c = 0; c < NCLS; ++c)
    if (gts[c] > 0.0f) { s += loss[c]; np += 1.0f; }
  out[0] = s / fmaxf(np, 1.0f);
}

extern "C" void kernel_launch(void* const* d_in, const int* in_sizes, int n_in,
                              void* d_out, int out_size, void* d_ws,
                              size_t ws_size, hipStream_t stream) {
  (void)in_sizes; (void)n_in; (void)out_size; (void)ws_size;
  const float* logits = (const float*)d_in[0];
  const int* labels = (const int*)d_in[1];

  float* W = (float*)d_ws;
  float* HIST = W;                               // 18 * NBINS floats
  float* SCC = W + (size_t)18 * NBINS;           // NBINS
  float* SCF = SCC + NBINS;                      // NBINS
  float* BSC = SCF + NBINS;                      // NSCAN_BLOCKS
  float* BSF = BSC + NSCAN_BLOCKS;
  float* BOC = BSF + NSCAN_BLOCKS;
  float* BOF = BOC + NSCAN_BLOCKS;
  float* GTS = BOF + NSCAN_BLOCKS;               // NCLS
  float* LOSS = GTS + NCLS;                      // NCLS

  lovasz_zero_kernel<<<2048, 256, 0, stream>>>(HIST, (long long)18 * NBINS,
                                               LOSS);
  lovasz_fill_kernel<<<NPIX / 256, 256, 0, stream>>>(logits, labels, HIST);

  for (int c = 0; c < NCLS; ++c) {
    const float* cnt = HIST + (size_t)c * 3 * NBINS;
    lovasz_scan_block_kernel<<<NSCAN_BLOCKS, 256, 0, stream>>>(cnt, SCC, BSC);
    lovasz_scan_block_kernel<<<NSCAN_BLOCKS, 256, 0, stream>>>(cnt + NBINS,
                                                               SCF, BSF);
    lovasz_scan_top_kernel<<<1, NSCAN_BLOCKS, 0, stream>>>(BSC, BSF, BOC, BOF,
                                                           GTS, c);
    lovasz_finalize_kernel<<<NBINS / 256, 256, 0, stream>>>(
        HIST, SCC, SCF, BOC, BOF, GTS, LOSS, c);
  }
  lovasz_combine_kernel<<<1, 1, 0, stream>>>(GTS, LOSS, (float*)d_out);
}